// LocalLinkage_78718160601452
// MI455X (gfx1250) — compile-verified
//
#include <hip/hip_runtime.h>
#include <stdint.h>

// Fused 3-layer LocallyConnected1D (K=2,S=2,C=F=1) for MI455X (gfx1250).
// out[b,q] = f(x[b,8q..8q+7]; W0[4q..4q+3,:], b0[4q..4q+3], W1[2q..2q+1,:], W2[q,:])
//
// Memory-bound streaming kernel: x staged into LDS via the CDNA5 Tensor Data
// Mover (tensor_load_to_lds, TH=NT + s_wait_tensorcnt), weights via contiguous
// per-thread float4 global loads (L2-resident across the batch dimension),
// non-temporal output stores.

typedef __attribute__((ext_vector_type(4))) unsigned int v4u;
typedef __attribute__((ext_vector_type(8))) int          v8i;
typedef __attribute__((ext_vector_type(4))) int          v4i;

#define TILE_Q 256                 // outputs per block
#define TILE_X (TILE_Q * 8)        // x elements per block (2048 floats = 8 KB)

__global__ __launch_bounds__(TILE_Q)
void lc1d_fused_kernel(const float* __restrict__ x,
                       const float* __restrict__ w0,
                       const float* __restrict__ b0,
                       const float* __restrict__ w1,
                       const float* __restrict__ w2,
                       float* __restrict__ out,
                       int L, int P2)
{
    __shared__ float tile[TILE_X];

    const int b   = blockIdx.x;          // batch (fast-varying -> weight reuse in L2)
    const int q0  = blockIdx.y * TILE_Q; // first output position of this block
    const int tid = threadIdx.x;

    const float* xsrc = x + (size_t)b * (size_t)L + (size_t)q0 * 8;

#if __has_builtin(__builtin_amdgcn_tensor_load_to_lds) && __has_builtin(__builtin_amdgcn_s_wait_tensorcnt)
    // ---- TDM: async DMA of the 8KB x tile into LDS (wave 0 issues & waits) ----
    if (tid < 32) {
        const unsigned lds_off = (unsigned)(uintptr_t)&tile[0]; // LDS byte offset (low 32b of shared ptr)
        const unsigned long long ga = (unsigned long long)(uintptr_t)xsrc;

        // D# group 0 (128b): count=1 | lds_addr | global_addr[56:0] | type=2
        v4u g0;
        g0[0] = 1u;                                   // count=1, user descriptor
        g0[1] = lds_off;                              // lds_addr (bytes)
        g0[2] = (unsigned)(ga & 0xFFFFFFFFull);       // global_addr[31:0]
        g0[3] = (unsigned)((ga >> 32) & 0x1FFFFFFull) // global_addr[56:32]
              | (2u << 30);                           // type = 2 ("image")

        // D# group 1 (256b): 1-D tensor, data_size=4B, tile = 2048 x 1
        v8i g1;
        g1[0] = (int)(2u << 16);                      // workgroup_mask=0, data_size=2 (4B)
        g1[1] = (int)(((unsigned)TILE_X & 0xFFFFu) << 16); // tensor_dim0[15:0] @ [31:16]
        g1[2] = (int)(((unsigned)TILE_X >> 16) & 0xFFFFu)  // tensor_dim0[31:16]
              | (1 << 16);                            // tensor_dim1 = 1
        g1[3] = (int)(((unsigned)TILE_X & 0xFFFFu) << 16); // tile_dim0 = 2048 @ [31:16]
        g1[4] = 1;                                    // tile_dim1 = 1, tile_dim2 = 0
        g1[5] = TILE_X;                               // tensor_dim0_stride[31:0]
        g1[6] = 0;                                    // stride hi / dim1_stride lo
        g1[7] = 0;                                    // dim1_stride hi

        v4i g2 = {0, 0, 0, 0};                        // <=2D tensor: groups 2/3 unused
        v4i g3 = {0, 0, 0, 0};
        v8i g4 = {0, 0, 0, 0, 0, 0, 0, 0};            // extra group (6-arg builtin form), unused for <=2D

        // cpol = 1 -> TH=NT: x is streamed exactly once into LDS; keep it out
        // of L2 so the (reused) weight working set stays resident there.
        __builtin_amdgcn_tensor_load_to_lds(g0, g1, g2, g3, g4, 1);
        __builtin_amdgcn_s_wait_tensorcnt(0);         // TENSORcnt==0: tile is in LDS
    }
#else
    // Fallback: cooperative vectorized copy (B128 per lane, fully coalesced)
    for (int i = tid * 4; i < TILE_X; i += TILE_Q * 4)
        *(float4*)&tile[i] = *(const float4*)&xsrc[i];
#endif
    __syncthreads();

    // ---- per-thread fused compute: one output element ----
    const int q = q0 + tid;

    const float4 xa = *(const float4*)&tile[tid * 8 + 0]; // x[8q..8q+3]
    const float4 xb = *(const float4*)&tile[tid * 8 + 4]; // x[8q+4..8q+7]

    const float4 w0a = *(const float4*)(w0 + (size_t)q * 8 + 0); // W0[4q..4q+1,:]
    const float4 w0b = *(const float4*)(w0 + (size_t)q * 8 + 4); // W0[4q+2..4q+3,:]
    const float4 bv  = *(const float4*)(b0 + (size_t)q * 4);     // b0[4q..4q+3]
    const float4 w1v = *(const float4*)(w1 + (size_t)q * 4);     // W1[2q..2q+1,:]
    const float2 w2v = *(const float2*)(w2 + (size_t)q * 2);     // W2[q,:]

    // Layer 0 (with bias)
    const float h00 = fmaf(xa.x, w0a.x, fmaf(xa.y, w0a.y, bv.x));
    const float h01 = fmaf(xa.z, w0a.z, fmaf(xa.w, w0a.w, bv.y));
    const float h02 = fmaf(xb.x, w0b.x, fmaf(xb.y, w0b.y, bv.z));
    const float h03 = fmaf(xb.z, w0b.z, fmaf(xb.w, w0b.w, bv.w));
    // Layer 1
    const float h10 = fmaf(h00, w1v.x, h01 * w1v.y);
    const float h11 = fmaf(h02, w1v.z, h03 * w1v.w);
    // Layer 2
    const float o   = fmaf(h10, w2v.x, h11 * w2v.y);

    // Write-once output: non-temporal store keeps L2 free for weights.
    __builtin_nontemporal_store(o, &out[(size_t)b * (size_t)P2 + (size_t)q]);
}

extern "C" void kernel_launch(void* const* d_in, const int* in_sizes, int n_in,
                              void* d_out, int out_size, void* d_ws, size_t ws_size,
                              hipStream_t stream)
{
    const float* x  = (const float*)d_in[0]; // [B, L, 1]
    const float* W0 = (const float*)d_in[1]; // [L/2, 2, 1]
    const float* b0 = (const float*)d_in[2]; // [L/2, 1]
    const float* W1 = (const float*)d_in[3]; // [L/4, 2, 1]
    const float* W2 = (const float*)d_in[4]; // [L/8, 2, 1]
    float* out = (float*)d_out;              // [B, L/8, 1]

    const int P2 = in_sizes[4] / 2;   // L/8  (= 32768)
    const int L  = P2 * 8;            // 262144
    const int B  = (int)(in_sizes[0] / L); // 256

    // batch fast-varying in grid.x -> 256 consecutive blocks share the same
    // 18KB weight slice -> weights stay in L2, HBM sees them ~once.
    dim3 grid(B, P2 / TILE_Q);
    lc1d_fused_kernel<<<grid, TILE_Q, 0, stream>>>(x, W0, b0, W1, W2, out, L, P2);
}